// Lesion_Location_Mining_65197603553367
// MI455X (gfx1250) — compile-verified
//
#include <hip/hip_runtime.h>
#include <hip/hip_bf16.h>
#include <math.h>

#define BATCH 8
#define CDIM  1024
#define NPIX  4096
#define TOPK  100
#define K2DIM 50
#define ROWS  112     // TOPK padded to 7 x 16 WMMA tiles

typedef __attribute__((ext_vector_type(2))) float v2f;
typedef __attribute__((ext_vector_type(8))) float v8f;

// ---------------------------------------------------------------------------
// Kernel 1: exact top-K (jax.lax.top_k semantics: value desc, index asc ties)
// One block per (side, batch). side 0 = fore (mask ch1), side 1 = back (ch0).
// ---------------------------------------------------------------------------
__global__ void topk_kernel(const float* __restrict__ soft_mask,
                            int* __restrict__ idx_ws) {
    const int s = blockIdx.x;
    const int b = blockIdx.y;
    const int ch = (s == 0) ? 1 : 0;
    __shared__ unsigned key[NPIX];
    __shared__ unsigned red[256];
    __shared__ unsigned bcast;
    const int tid = threadIdx.x;
    const float* src = soft_mask + (size_t)(b * 2 + ch) * NPIX;
    for (int i = tid; i < NPIX; i += 256) {
        unsigned u = __float_as_uint(src[i]);
        u = (u & 0x80000000u) ? ~u : (u | 0x80000000u);  // monotone float->uint
        key[i] = u;
    }
    __syncthreads();
    // binary search for V = K-th largest key (max x with count(key>=x) >= K)
    unsigned lo = 0u;
    for (int bit = 31; bit >= 0; --bit) {
        unsigned mid = lo | (1u << bit);
        unsigned cnt = 0;
        for (int i = tid; i < NPIX; i += 256) cnt += (key[i] >= mid) ? 1u : 0u;
        red[tid] = cnt;
        __syncthreads();
        for (int off = 128; off > 0; off >>= 1) {
            if (tid < off) red[tid] += red[tid + off];
            __syncthreads();
        }
        if (tid == 0) bcast = red[0];
        __syncthreads();
        if (bcast >= (unsigned)TOPK) lo = mid;
        __syncthreads();
    }
    const unsigned V = lo;
    int* idx = idx_ws + (b * 2 + s) * 128;
    for (int i = tid; i < NPIX; i += 256) {
        unsigned ki = key[i];
        if (ki >= V) {
            int rank = 0;
            for (int j = 0; j < NPIX; ++j) {
                unsigned kj = key[j];
                rank += ((kj > ki) || (kj == ki && j < i)) ? 1 : 0;
            }
            if (rank < TOPK) idx[rank] = i;   // ranks 0..K-1 each hit once
        }
    }
}

// ---------------------------------------------------------------------------
// Kernel 2: gather protos, gate MLP, write gated protos (112 padded rows,
// row-major [k][c]) + proto L2 norms. One block per (side, batch).
// ---------------------------------------------------------------------------
__global__ void gate_kernel(const float* __restrict__ feats,
                            const float* __restrict__ convw_f,
                            const float* __restrict__ fc1w_f,
                            const float* __restrict__ fc1b_f,
                            const float* __restrict__ fc2w_f,
                            const float* __restrict__ fc2b_f,
                            const float* __restrict__ convw_b,
                            const float* __restrict__ fc1w_b,
                            const float* __restrict__ fc1b_b,
                            const float* __restrict__ fc2w_b,
                            const float* __restrict__ fc2b_b,
                            const int* __restrict__ idx_ws,
                            float* __restrict__ pg_ws,
                            float* __restrict__ np_ws) {
    const int s = blockIdx.x;
    const int b = blockIdx.y;
    const float* convw = s ? convw_b : convw_f;
    const float* fc1w  = s ? fc1w_b  : fc1w_f;
    const float* fc1b  = s ? fc1b_b  : fc1b_f;
    const float* fc2w  = s ? fc2w_b  : fc2w_f;
    const float* fc2b  = s ? fc2b_b  : fc2b_f;
    __shared__ float cw[CDIM];
    __shared__ float pw[TOPK];
    __shared__ float x1[K2DIM];
    __shared__ float g[TOPK];
    __shared__ int   sidx[TOPK];
    const int tid = threadIdx.x;
    const int* idx = idx_ws + (b * 2 + s) * 128;
    if (tid < TOPK) sidx[tid] = idx[tid];
    for (int i = tid; i < CDIM; i += 256) cw[i] = convw[i];
    __syncthreads();
    const float* fb = feats + (size_t)b * CDIM * NPIX;
    if (tid < TOPK) {
        const int n = sidx[tid];
        float acc = 0.f;
        for (int c = 0; c < CDIM; ++c) acc += fb[(size_t)c * NPIX + n] * cw[c];
        pw[tid] = fmaxf(acc, 0.f);           // relu (relu(relu(x)) == relu(x))
    }
    __syncthreads();
    if (tid < K2DIM) {
        float acc = fc1b[tid];
        for (int k = 0; k < TOPK; ++k) acc += pw[k] * fc1w[tid * TOPK + k];
        x1[tid] = acc;
    }
    __syncthreads();
    if (tid < TOPK) {
        float acc = fc2b[tid];
        for (int j = 0; j < K2DIM; ++j) acc += x1[j] * fc2w[tid * K2DIM + j];
        g[tid] = 1.f / (1.f + expf(-acc));   // sigmoid
    }
    __syncthreads();
    float* pg = pg_ws + (size_t)(b * 2 + s) * ROWS * CDIM;
    if (tid < ROWS) {
        const bool live = (tid < TOPK);
        const float gk = live ? g[tid] : 0.f;
        const int   n  = live ? sidx[tid] : 0;
        float n2 = 0.f;
        for (int c = 0; c < CDIM; ++c) {
            float v = live ? fb[(size_t)c * NPIX + n] * gk : 0.f;
            pg[(size_t)tid * CDIM + c] = v;
            n2 += v * v;
        }
        np_ws[(b * 2 + s) * ROWS + tid] = sqrtf(n2);
    }
}

// ---------------------------------------------------------------------------
// Kernel 3: WMMA fp32 cosine attention.
// att[k,n] = sum_c pg[k,c] * other[c,n];  other = feats * hard-mask.
// out_att[n] = relu( max_k att[k,n] / (||pg_k|| * ||other_n|| + 1e-8) ).
// One block (8 waves) per (128-col block, side, batch). 7 M-tiles per wave.
// Inner loop: batch 16 ds_load_2addr (8-wide K slab), then 14 back-to-back
// WMMAs -> one dscnt wait per 14 matrix ops instead of one per op.
// ---------------------------------------------------------------------------
#define NT   128
#define CCH  128
#define ASTR 132
#define BSTR 133
#define SMEM_FLOATS (ROWS*ASTR + NT*BSTR + NT + 256 + NT + ROWS)

__global__ void attn_kernel(const float* __restrict__ feats,
                            const float* __restrict__ soft_mask,
                            const float* __restrict__ pg_ws,
                            const float* __restrict__ np_ws,
                            float* __restrict__ att_ws) {
    const int nb = blockIdx.x;   // 0..31 column blocks
    const int s  = blockIdx.y;   // 0 fore (other = bg), 1 back (other = fg)
    const int b  = blockIdx.z;
    const int tid  = threadIdx.x;
    const int lane = tid & 31;
    const int wv   = tid >> 5;   // 0..7, each wave owns 16 columns

    extern __shared__ float smem[];
    float* As    = smem;               // [ROWS][ASTR]  A = gated protos (k x c)
    float* Bs    = As + ROWS * ASTR;   // [NT][BSTR]    B stored transposed [n][c]
    float* maskv = Bs + NT * BSTR;     // [NT]
    float* red   = maskv + NT;         // [256]
    float* nfv   = red + 256;          // [NT]
    float* npv   = nfv + NT;           // [ROWS]

    const int n0 = nb * NT;
    if (tid < NT) {
        int gn = n0 + tid;
        float s0 = soft_mask[(size_t)(b * 2 + 0) * NPIX + gn];
        float s1 = soft_mask[(size_t)(b * 2 + 1) * NPIX + gn];
        float fg = (s1 > s0) ? 1.f : 0.f;            // argmax tie -> index 0
        maskv[tid] = (s == 0) ? (1.f - fg) : fg;
    }
    if (tid < ROWS) npv[tid] = np_ws[(b * 2 + s) * ROWS + tid];

    const float* pg = pg_ws + (size_t)(b * 2 + s) * ROWS * CDIM;
    const float* fb = feats + (size_t)b * CDIM * NPIX;

    v8f acc[7] = {};
    float nfloc = 0.f;
    const int myn   = tid & (NT - 1);   // column owned for B loads / nf accum
    const int crow0 = tid >> 7;         // 0/1: c-row parity

    const int h2 = (lane >> 4) * 2;     // K-pair select for 16x16x4 f32 frags
    const int nn = lane & 15;           // M row (A) == N col (B) within tile
    const int col = wv * 16 + nn;

    for (int c0 = 0; c0 < CDIM; c0 += CCH) {
        __syncthreads();
        // stage A chunk: 112 rows x 128 c (coalesced along c)
        for (int i = tid; i < ROWS * CCH; i += 256) {
            int r = i >> 7, cc = i & (CCH - 1);
            As[r * ASTR + cc] = pg[(size_t)r * CDIM + c0 + cc];
        }
        // stage B chunk transposed [n][c], masked; accumulate ||other_n||^2
        float mv = maskv[myn];
        for (int cc = crow0; cc < CCH; cc += 2) {
            float v = fb[(size_t)(c0 + cc) * NPIX + n0 + myn] * mv;
            Bs[myn * BSTR + cc] = v;
            nfloc += v * v;
        }
        __syncthreads();

        // prefetch next chunk's global lines (overlaps with WMMA work below)
        if (c0 + CCH < CDIM) {
            const float* pgn = pg + (size_t)(c0 + CCH);
            for (int l = tid; l < 448; l += 256)              // A: 112 rows x 4 lines
                __builtin_prefetch(pgn + (size_t)(l >> 2) * CDIM + (l & 3) * 32, 0, 0);
            const float* fbn = fb + (size_t)(c0 + CCH) * NPIX + n0;
            for (int l = tid; l < 512; l += 256)              // B: 128 rows x 4 lines
                __builtin_prefetch(fbn + (size_t)(l >> 2) * NPIX + (l & 3) * 32, 0, 0);
        }

        for (int kk = 0; kk < CCH; kk += 8) {
            // batch-load all fragments for an 8-wide K slab
            v2f bf0, bf1;
            const int bOff = col * BSTR + kk + h2;
            bf0.x = Bs[bOff];     bf0.y = Bs[bOff + 1];
            bf1.x = Bs[bOff + 4]; bf1.y = Bs[bOff + 5];
            v2f af0[7], af1[7];
#pragma unroll
            for (int mt = 0; mt < 7; ++mt) {
                const int aOff = (mt * 16 + nn) * ASTR + kk + h2;
                af0[mt].x = As[aOff];     af0[mt].y = As[aOff + 1];
                af1[mt].x = As[aOff + 4]; af1[mt].y = As[aOff + 5];
            }
            // 14 back-to-back WMMAs (independent D regs; C-chains pipeline)
#pragma unroll
            for (int mt = 0; mt < 7; ++mt)
                acc[mt] = __builtin_amdgcn_wmma_f32_16x16x4_f32(
                    false, af0[mt], false, bf0, (short)0, acc[mt], false, false);
#pragma unroll
            for (int mt = 0; mt < 7; ++mt)
                acc[mt] = __builtin_amdgcn_wmma_f32_16x16x4_f32(
                    false, af1[mt], false, bf1, (short)0, acc[mt], false, false);
        }
    }
    // finalize column norms (two partial owners per column)
    red[tid] = nfloc;
    __syncthreads();
    if (tid < NT) nfv[tid] = sqrtf(red[tid] + red[tid + NT]);
    __syncthreads();

    // epilogue: cosine divide, max over all 112 rows, relu, store
    const int h = lane >> 4;
    const float nfc = nfv[col];
    float vmax = -1e30f;
#pragma unroll
    for (int mt = 0; mt < 7; ++mt) {
#pragma unroll
        for (int r = 0; r < 8; ++r) {
            int m = mt * 16 + r + 8 * h;           // C/D layout: VGPR r, half h
            float val = acc[mt][r] / (npv[m] * nfc + 1e-8f);
            vmax = fmaxf(vmax, val);
        }
    }
    float oth = __shfl_xor(vmax, 16, 32);          // combine the two M halves
    vmax = fmaxf(vmax, oth);
    if (h == 0) {
        att_ws[(size_t)(s * BATCH + b) * NPIX + n0 + col] = fmaxf(vmax, 0.f);
    }
}

// ---------------------------------------------------------------------------
// Kernel 4: out = feats * (1 + fg_soft - back_att + fore_att)   (float4)
// ---------------------------------------------------------------------------
__global__ void combine_kernel(const float* __restrict__ feats,
                               const float* __restrict__ soft_mask,
                               const float* __restrict__ att_ws,
                               float* __restrict__ out) {
    const size_t g = (size_t)blockIdx.x * blockDim.x + threadIdx.x;
    const size_t total = (size_t)BATCH * CDIM * (NPIX / 4);
    if (g >= total) return;
    const int n4   = (int)(g % (NPIX / 4));
    const size_t r = g / (NPIX / 4);
    const int c = (int)(r % CDIM);
    const int b = (int)(r / CDIM);
    const float4 f  = ((const float4*)(feats + ((size_t)b * CDIM + c) * NPIX))[n4];
    const float4 sm = ((const float4*)(soft_mask + (size_t)(b * 2 + 1) * NPIX))[n4];
    const float4 fo = ((const float4*)(att_ws + (size_t)(0 * BATCH + b) * NPIX))[n4];
    const float4 ba = ((const float4*)(att_ws + (size_t)(1 * BATCH + b) * NPIX))[n4];
    float4 o;
    o.x = f.x * (1.f + sm.x - ba.x + fo.x);
    o.y = f.y * (1.f + sm.y - ba.y + fo.y);
    o.z = f.z * (1.f + sm.z - ba.z + fo.z);
    o.w = f.w * (1.f + sm.w - ba.w + fo.w);
    ((float4*)(out + ((size_t)b * CDIM + c) * NPIX))[n4] = o;
}

// ---------------------------------------------------------------------------
extern "C" void kernel_launch(void* const* d_in, const int* in_sizes, int n_in,
                              void* d_out, int out_size, void* d_ws, size_t ws_size,
                              hipStream_t stream) {
    const float* feats     = (const float*)d_in[0];
    const float* soft_mask = (const float*)d_in[1];
    const float* convw_f   = (const float*)d_in[2];
    const float* fc1w_f    = (const float*)d_in[3];
    const float* fc1b_f    = (const float*)d_in[4];
    const float* fc2w_f    = (const float*)d_in[5];
    const float* fc2b_f    = (const float*)d_in[6];
    const float* convw_b   = (const float*)d_in[7];
    const float* fc1w_b    = (const float*)d_in[8];
    const float* fc1b_b    = (const float*)d_in[9];
    const float* fc2w_b    = (const float*)d_in[10];
    const float* fc2b_b    = (const float*)d_in[11];
    float* out = (float*)d_out;

    // workspace layout (floats): [idx 2048 ints][pg 16*112*1024][np 16*112][att 2*8*4096]
    int*   idx_ws = (int*)d_ws;
    float* pg_ws  = (float*)d_ws + 2048;
    float* np_ws  = pg_ws + (size_t)16 * ROWS * CDIM;
    float* att_ws = np_ws + 16 * ROWS;

    topk_kernel<<<dim3(2, BATCH), 256, 0, stream>>>(soft_mask, idx_ws);
    gate_kernel<<<dim3(2, BATCH), 256, 0, stream>>>(
        feats, convw_f, fc1w_f, fc1b_f, fc2w_f, fc2b_f,
        convw_b, fc1w_b, fc1b_b, fc2w_b, fc2b_b,
        idx_ws, pg_ws, np_ws);
    attn_kernel<<<dim3(NPIX / NT, 2, BATCH), 256, SMEM_FLOATS * sizeof(float), stream>>>(
        feats, soft_mask, pg_ws, np_ws, att_ws);
    const size_t tot = (size_t)BATCH * CDIM * (NPIX / 4);
    combine_kernel<<<(unsigned)((tot + 255) / 256), 256, 0, stream>>>(
        feats, soft_mask, att_ws, out);
}